// TopKRouter_32315333935433
// MI455X (gfx1250) — compile-verified
//
#include <hip/hip_runtime.h>

typedef __attribute__((ext_vector_type(2))) float v2f;
typedef __attribute__((ext_vector_type(8))) float v8f;
typedef __attribute__((ext_vector_type(4))) int   v4i;

#define HIDDEN 2048
#define NEXP   64
#define TPB    128   // tokens per block
#define TPW    16    // tokens per wave
#define KC     32    // K-chunk staged in LDS
#define NCHUNK (HIDDEN / KC)
#define WROW   80    // padded LDS row stride (dwords); 2 rows = 160 ≡ 32 (mod 64)
                     // -> half-wave groups read disjoint bank sets: conflict-free

#if __has_builtin(__builtin_amdgcn_global_load_async_to_lds_b128) && \
    __has_builtin(__builtin_amdgcn_s_wait_asynccnt)
#define USE_ASYNC 1
#define WAIT_ASYNC(n) __builtin_amdgcn_s_wait_asynccnt(n)
#else
#define USE_ASYNC 0
#define WAIT_ASYNC(n)
#endif

typedef __attribute__((address_space(1))) v4i glob_v4i;
typedef __attribute__((address_space(3))) v4i lds_v4i;

__device__ __forceinline__ void cp16_to_lds(const float* g, float* l)
{
#if USE_ASYNC
    // async global -> LDS, 16B per lane, tracked by ASYNCcnt
    __builtin_amdgcn_global_load_async_to_lds_b128(
        (glob_v4i*)g, (lds_v4i*)l, /*offset=*/0, /*cpol=*/0);
#else
    *(float4*)l = *(const float4*)g;
#endif
}

// ---------------------------------------------------------------------------
// Fused: logits GEMM (fp32 WMMA, async-LDS staged B) -> softmax -> top-2 -> aux
// ---------------------------------------------------------------------------
__global__ __launch_bounds__(256) void router_gemm_topk(
    const float* __restrict__ x, const float* __restrict__ Wg,
    float* __restrict__ out, float* __restrict__ psum, float* __restrict__ pcnt,
    int nTokens)
{
    __shared__ float s_logit[TPB][NEXP + 1];   // +1 pad: conflict-free banks
    __shared__ float s_w[2][KC * WROW];        // double-buffered W_gate chunks
    __shared__ float s_max[TPB];
    __shared__ float s_rsum[TPB];
    __shared__ int   s_cnt[NEXP];

    const int tid  = threadIdx.x;
    const int wave = tid >> 5;
    const int lane = tid & 31;
    const int lg   = lane >> 4;   // half-wave group (0/1) -> K pair select
    const int lm   = lane & 15;   // M row (A) / N col (B)
    const int blk_t0  = blockIdx.x * TPB;
    const int wave_t0 = blk_t0 + wave * TPW;

    // Stage one KC x 64 chunk of W_gate into LDS (coalesced b128 per lane).
    // KC*64 = 2048 dwords / (256 thr * 4 dw) = 2 copies per thread.
    auto stage = [&](int chunk, int buf) {
        const float* gbase = Wg + (size_t)chunk * KC * NEXP;
        #pragma unroll
        for (int s2 = 0; s2 < (KC * NEXP) / (256 * 4); ++s2) {
            const int idx = (s2 * 256 + tid) * 4;   // dword idx in chunk
            const int r = idx >> 6, c = idx & 63;   // 16B stays within a row
            cp16_to_lds(gbase + idx, &s_w[buf][r * WROW + c]);
        }
    };

    // ---- Phase 1: 16x64 logit tile per wave via V_WMMA_F32_16X16X4_F32 ----
    // A layout (16x4 f32): lane lm = M row; v0/v1 = K 2*lg, 2*lg+1
    const float* aptr = x + (size_t)(wave_t0 + lm) * HIDDEN + (lg << 1);
    v8f c0 = {}, c1 = {}, c2 = {}, c3 = {};

    stage(0, 0);
    for (int ch = 0; ch < NCHUNK; ++ch) {
        if (ch + 1 < NCHUNK) {            // overlap next chunk's copy
            stage(ch + 1, (ch + 1) & 1);
            WAIT_ASYNC(2);                // chunk ch done; next still in flight
        } else {
            WAIT_ASYNC(0);
        }
        __syncthreads();

        // B layout (4x16 f32): lane lm = N; v0/v1 = K rows 2*lg, 2*lg+1
        const float* wb = &s_w[ch & 1][(lg << 1) * WROW + lm];
        #pragma unroll
        for (int s = 0; s < KC / 4; ++s) {
            v2f a = *(const v2f*)aptr;    // global_load_b64, the HBM stream
            aptr += 4;
            v2f b0 = { wb[ 0], wb[WROW +  0] };   // -> ds_load_2addr_b32
            v2f b1 = { wb[16], wb[WROW + 16] };
            v2f b2 = { wb[32], wb[WROW + 32] };
            v2f b3 = { wb[48], wb[WROW + 48] };
            wb += 4 * WROW;
            c0 = __builtin_amdgcn_wmma_f32_16x16x4_f32(false, a, false, b0, (short)0, c0, false, false);
            c1 = __builtin_amdgcn_wmma_f32_16x16x4_f32(false, a, false, b1, (short)0, c1, false, false);
            c2 = __builtin_amdgcn_wmma_f32_16x16x4_f32(false, a, false, b2, (short)0, c2, false, false);
            c3 = __builtin_amdgcn_wmma_f32_16x16x4_f32(false, a, false, b3, (short)0, c3, false, false);
        }
        __syncthreads();                  // buffer free before reuse
    }

    // C/D layout: VGPR r, lane -> M = r + 8*lg, N = lm + 16*tile
    {
        const int mrow = wave * TPW + 8 * lg;
        #pragma unroll
        for (int r = 0; r < 8; ++r) {
            s_logit[mrow + r][lm +  0] = c0[r];
            s_logit[mrow + r][lm + 16] = c1[r];
            s_logit[mrow + r][lm + 32] = c2[r];
            s_logit[mrow + r][lm + 48] = c3[r];
        }
    }
    __syncthreads();
    if (tid < NEXP) s_cnt[tid] = 0;
    __syncthreads();

    // ---- Phase 2: per-token softmax + top-2 (one token per thread) ----
    if (tid < TPB) {
        const int token = blk_t0 + tid;
        const float* row = s_logit[tid];
        float m = row[0];
        #pragma unroll
        for (int e = 1; e < NEXP; ++e) m = fmaxf(m, row[e]);
        float s = 0.f;
        #pragma unroll
        for (int e = 0; e < NEXP; ++e) s += __expf(row[e] - m);
        const float rs = 1.f / s;

        // top-2 on logits (monotonic under softmax); ascending scan with
        // strict '>' reproduces lax.top_k first-index tie-breaking.
        int i1 = 0; float v1 = row[0];
        #pragma unroll
        for (int e = 1; e < NEXP; ++e) { float v = row[e]; if (v > v1) { v1 = v; i1 = e; } }
        int i2 = (i1 == 0) ? 1 : 0; float v2 = row[i2];
        #pragma unroll
        for (int e = 0; e < NEXP; ++e) {
            if (e == i1) continue;
            float v = row[e]; if (v > v2) { v2 = v; i2 = e; }
        }
        const float w1 = __expf(v1 - m) * rs;
        const float w2 = __expf(v2 - m) * rs;
        const float d  = w1 + w2 + 1e-9f;
        out[(size_t)token * 2 + 0] = w1 / d;
        out[(size_t)token * 2 + 1] = w2 / d;
        float* oidx = out + (size_t)nTokens * 2;
        oidx[(size_t)token * 2 + 0] = (float)i1;
        oidx[(size_t)token * 2 + 1] = (float)i2;

        s_max[tid]  = m;
        s_rsum[tid] = rs;
        atomicAdd(&s_cnt[i1], 1);
    }
    __syncthreads();

    // ---- Phase 3: transposed per-expert prob sums; 64 atomics per block ----
    if (tid < NEXP) {
        float acc = 0.f;
        #pragma unroll 8
        for (int t = 0; t < TPB; ++t)
            acc += __expf(s_logit[t][tid] - s_max[t]) * s_rsum[t];
        atomicAdd(&psum[tid], acc);
        atomicAdd(&pcnt[tid], (float)s_cnt[tid]);
    }
}

// ---------------------------------------------------------------------------
__global__ void router_zero_ws(float* ws)
{
    int i = blockIdx.x * blockDim.x + threadIdx.x;
    if (i < 2 * NEXP) ws[i] = 0.f;
}

__global__ void router_finalize(const float* __restrict__ psum,
                                const float* __restrict__ pcnt,
                                float* __restrict__ aux, int nTokens)
{
    __shared__ float red[NEXP];
    const int e = threadIdx.x;
    const float inv = 1.f / (float)nTokens;
    red[e] = (pcnt[e] * inv) * (psum[e] * inv);
    __syncthreads();
    for (int s = NEXP / 2; s > 0; s >>= 1) {
        if (e < s) red[e] += red[e + s];
        __syncthreads();
    }
    if (e == 0) aux[0] = (float)NEXP * red[0];
}

// ---------------------------------------------------------------------------
extern "C" void kernel_launch(void* const* d_in, const int* in_sizes, int n_in,
                              void* d_out, int out_size, void* d_ws, size_t ws_size,
                              hipStream_t stream)
{
    const float* x  = (const float*)d_in[0];   // (nTokens, HIDDEN) f32
    const float* Wg = (const float*)d_in[1];   // (HIDDEN, NEXP)   f32
    float* out  = (float*)d_out;               // [w(2T) | idx(2T) | aux(1)]
    float* psum = (float*)d_ws;                // [NEXP] prob sums
    float* pcnt = psum + NEXP;                 // [NEXP] top-1 counts

    const int nTokens = in_sizes[0] / HIDDEN;

    router_zero_ws<<<1, 128, 0, stream>>>(psum);
    router_gemm_topk<<<nTokens / TPB, 256, 0, stream>>>(x, Wg, out, psum, pcnt, nTokens);
    router_finalize<<<1, NEXP, 0, stream>>>(psum, pcnt, out + (size_t)nTokens * 4, nTokens);
}